// YoloLoss_77541339562327
// MI455X (gfx1250) — compile-verified
//
#include <hip/hip_runtime.h>
#include <hip/hip_bf16.h>
#include <math.h>

#define BB 2
#define AA 3
#define SS 8
#define CCH 20
#define IMG 256
#define NCELL (BB*AA*SS*SS)     // 384
#define NPIX  (BB*IMG*IMG)      // 131072
#define SENT  1e30f             // sentinel: unpainted / never-maskable

typedef __attribute__((ext_vector_type(2))) float v2f;
typedef __attribute__((ext_vector_type(8))) float v8f;

__device__ __forceinline__ float sigf(float x) { return 1.0f / (1.0f + expf(-x)); }

// _axis_mask endpoint: trunc(clip(x,±1e9)) -> int, negative wraps from L, clamp to [0,L]
__device__ __forceinline__ int axis_clip(float x) {
    x = fminf(fmaxf(x, -1e9f), 1e9f);
    int t = (int)truncf(x);
    if (t < 0) { t = IMG + t; if (t < 0) t = 0; }
    else if (t > IMG) t = IMG;
    return t;
}

// ---------------------------------------------------------------------------
// Kernel 1: detection losses. 384 cells -> 6 partial sums; final reduction of
// the 4 tree-reduced chunks via V_WMMA_F32_16X16X4_F32 (A rows = quantities,
// B = ones => D rows hold the 6 totals in full f32).
// ws[0]=sum objf, ws[1]=sum noobj, ws[2]=bce num, ws[3]=obj num,
// ws[4]=box num, ws[5]=class num
// ---------------------------------------------------------------------------
__global__ __launch_bounds__(512) void det_loss_kernel(
    const float* __restrict__ pred, const float* __restrict__ tgt,
    const float* __restrict__ anchors, float* __restrict__ ws)
{
    __shared__ float red[6][512];
    const int tid = threadIdx.x;
    float q0 = 0.f, q1 = 0.f, q2 = 0.f, q3 = 0.f, q4 = 0.f, q5 = 0.f;

    if (tid < NCELL) {
        // linear index tid == ((b*A + a)*S + i)*S + j  (memory layout order)
        const int a = (tid / (SS * SS)) % AA;
        const float* p = pred + tid * (6 + CCH);
        const float* t = tgt + tid * 6;
        const float aw = anchors[2 * a], ah = anchors[2 * a + 1];

        const float p0 = p[0], p1 = p[1], p2 = p[2], p3 = p[3], p4 = p[4];
        const float t0 = t[0], t1 = t[1], t2 = t[2], t3 = t[3], t4 = t[4], t5 = t[5];
        const float objf  = (t0 == 1.0f) ? 1.0f : 0.0f;
        const float noobj = (t0 == 0.0f) ? 1.0f : 0.0f;

        q0 = objf;
        q1 = noobj;
        // masked BCE-with-logits numerator
        q2 = (fmaxf(p0, 0.0f) - p0 * t0 + log1pf(expf(-fabsf(p0)))) * noobj;

        // IoU (midpoint) between decoded box and target box
        const float bx = sigf(p1), by = sigf(p2);
        const float bw = expf(p3) * aw, bh = expf(p4) * ah;
        const float ax1 = bx - bw * 0.5f, ax2 = bx + bw * 0.5f;
        const float ay1 = by - bh * 0.5f, ay2 = by + bh * 0.5f;
        const float bx1 = t1 - t3 * 0.5f, bx2 = t1 + t3 * 0.5f;
        const float by1 = t2 - t4 * 0.5f, by2 = t2 + t4 * 0.5f;
        const float iw = fmaxf(fminf(ax2, bx2) - fmaxf(ax1, bx1), 0.0f);
        const float ih = fmaxf(fminf(ay2, by2) - fmaxf(ay1, by1), 0.0f);
        const float inter = iw * ih;
        const float areaA = fabsf((ax2 - ax1) * (ay2 - ay1));
        const float areaB = fabsf((bx2 - bx1) * (by2 - by1));
        const float iou = inter / (areaA + areaB - inter + 1e-6f);
        const float d0 = sigf(p0) - iou * t0;
        q3 = d0 * d0 * objf;

        // box loss numerator
        const float e1 = bx - t1, e2 = by - t2;
        const float e3 = p3 - logf(1e-16f + t3 / aw);
        const float e4 = p4 - logf(1e-16f + t4 / ah);
        q4 = (e1 * e1 + e2 * e2 + e3 * e3 + e4 * e4) * objf;

        // class NLL numerator (log-softmax over 20 logits)
        float m = p[6];
        #pragma unroll
        for (int c = 1; c < CCH; ++c) m = fmaxf(m, p[6 + c]);
        float se = 0.0f;
        #pragma unroll
        for (int c = 0; c < CCH; ++c) se += expf(p[6 + c] - m);
        const float lse = m + logf(se);
        const int lab = (int)t5;
        q5 = (lse - p[6 + lab]) * objf;
    }

    red[0][tid] = q0; red[1][tid] = q1; red[2][tid] = q2;
    red[3][tid] = q3; red[4][tid] = q4; red[5][tid] = q5;
    __syncthreads();

    // fixed-order tree reduce 512 -> 4 chunks per quantity (deterministic)
    for (int s = 256; s >= 4; s >>= 1) {
        if (tid < s) {
            #pragma unroll
            for (int qi = 0; qi < 6; ++qi) red[qi][tid] += red[qi][tid + s];
        }
        __syncthreads();
    }

    // Wave 0 (EXEC all ones): WMMA row-sum of the remaining 4 chunks.
    // A(16x4): lanes 0-15 -> (M=lane, K=0/1 in v0/v1); lanes 16-31 -> (M=lane-16, K=2/3)
    // B = ones  =>  D[M][N] = sum_K A[M][K]; lane 0 holds totals in d[0..5].
    if (tid < 32) {
        const int m = tid & 15;
        const int kb = (tid < 16) ? 0 : 2;
        v2f av;
        av.x = (m < 6) ? red[m][kb]     : 0.0f;
        av.y = (m < 6) ? red[m][kb + 1] : 0.0f;
        v2f bv; bv.x = 1.0f; bv.y = 1.0f;
        v8f cz = {};
        v8f dres = __builtin_amdgcn_wmma_f32_16x16x4_f32(
            false, av, false, bv, (short)0, cz, false, false);
        if (tid == 0) {
            #pragma unroll
            for (int qi = 0; qi < 6; ++qi) ws[qi] = dres[qi];
        }
    }
}

// ---------------------------------------------------------------------------
// Kernel 2: sequential 384-step depth scan. Incremental SILog statistics:
// per map we keep (N, Sum g, Sum g^2); each pixel stores its CURRENT g
// contribution (SENT = unpainted), and log(depth_target) is precomputed
// (SENT = tgt<1, never maskable). Inner loop: 2 loads + 1 store, no logs.
// ---------------------------------------------------------------------------
__global__ __launch_bounds__(1024) void depth_loss_kernel(
    const float* __restrict__ pred, const float* __restrict__ anchors,
    const float* __restrict__ dtgt, float* __restrict__ gp, float* __restrict__ gs,
    float* __restrict__ ltg, const float* __restrict__ det, float* __restrict__ out)
{
    __shared__ double sSg[2], sSg2[2];
    __shared__ unsigned sN[2];
    __shared__ double sLoss;
    __shared__ float sCnt;
    __shared__ double wSg[2][32], wSg2[2][32];
    __shared__ unsigned wN[2][32];

    const int tid = threadIdx.x;
    const int nthr = blockDim.x;

    // fresh state every launch (harness does not re-poison; no cross-call state)
    for (int i = tid; i < NPIX; i += nthr) {
        gp[i] = SENT; gs[i] = SENT;
        const float tg = dtgt[i];
        ltg[i] = (tg >= 1.0f) ? logf(tg) : SENT;
    }
    if (tid == 0) {
        sSg[0] = sSg[1] = 0.0; sSg2[0] = sSg2[1] = 0.0;
        sN[0] = sN[1] = 0u; sLoss = 0.0; sCnt = 0.0f;
    }
    __syncthreads();

    #pragma unroll 1
    for (int step = 0; step < NCELL; ++step) {
        // step = ((k*S + i)*S + j)*B + b
        const int k = step >> 7;
        const int i = (step >> 4) & 7;
        const int j = (step >> 1) & 7;
        const int b = step & 1;

        const float* p = pred + (((b * AA + k) * SS + i) * SS + j) * (6 + CCH);
        const float aw = anchors[2 * k], ah = anchors[2 * k + 1];

        // slice k: sigmoid applied (k+2) times, exp()*anchor applied (k+1) times
        float sx = p[1], sy = p[2];
        for (int n = 0; n < k + 2; ++n) { sx = sigf(sx); sy = sigf(sy); }
        float w = p[3], h = p[4];
        for (int n = 0; n < k + 1; ++n) { w = expf(w) * aw; h = expf(h) * ah; }

        const float cx = (sx + (float)j) * 32.0f;   // /S*IMG with S=8,IMG=256
        const float cy = (sy + (float)i) * 32.0f;
        w *= 32.0f; h *= 32.0f;
        const bool valid = (w > 0.0f) && (h > 0.0f) && isfinite(w) && isfinite(h);
        if (!valid) continue;   // uniform: ref paints nothing, adds nothing

        const float val = fmaxf(1.0f, p[5]);
        const float logval = logf(val);

        unsigned dN[2] = {0u, 0u};
        float fSg[2] = {0.f, 0.f}, fSg2[2] = {0.f, 0.f};

        #pragma unroll 1
        for (int rsel = 0; rsel < 2; ++rsel) {
            const float hw = (rsel == 0) ? w * 0.5f : w * 0.25f;
            const float hh = (rsel == 0) ? h * 0.5f : h * 0.25f;
            // NB: reference applies the cx-range to the FIRST image dim (rows)
            const int r0 = axis_clip(cx - hw), r1 = axis_clip(cx + hw);
            const int c0 = axis_clip(cy - hh), c1 = axis_clip(cy + hh);
            const int nr = r1 - r0, nc = c1 - c0;
            if (nr <= 0 || nc <= 0) continue;

            float* arr = (rsel == 0) ? gp : gs;
            const int area = nr * nc;
            const int base = b * IMG * IMG;
            for (int t = tid; t < area; t += nthr) {
                const int px = base + (r0 + t / nc) * IMG + (c0 + t % nc);
                const float lt = ltg[px];
                if (lt < SENT) {                 // tgt >= 1: pixel is maskable
                    const float gn = logval - lt;
                    const float og = arr[px];
                    arr[px] = gn;
                    if (og < SENT) {             // already masked: swap contribution
                        fSg[rsel]  += gn - og;
                        fSg2[rsel] += gn * gn - og * og;
                    } else {                     // newly masked pixel
                        dN[rsel] += 1u;
                        fSg[rsel]  += gn;
                        fSg2[rsel] += gn * gn;
                    }
                }
            }
        }

        // one combined deterministic block reduction for both maps
        double rSg[2], rSg2[2];
        rSg[0] = (double)fSg[0]; rSg[1] = (double)fSg[1];
        rSg2[0] = (double)fSg2[0]; rSg2[1] = (double)fSg2[1];
        for (int off = 16; off > 0; off >>= 1) {
            #pragma unroll
            for (int rsel = 0; rsel < 2; ++rsel) {
                rSg[rsel]  += __shfl_down(rSg[rsel],  off);
                rSg2[rsel] += __shfl_down(rSg2[rsel], off);
                dN[rsel]   += __shfl_down(dN[rsel],   off);
            }
        }
        const int lane = tid & 31, wid = tid >> 5;
        if (lane == 0) {
            #pragma unroll
            for (int rsel = 0; rsel < 2; ++rsel) {
                wSg[rsel][wid] = rSg[rsel]; wSg2[rsel][wid] = rSg2[rsel]; wN[rsel][wid] = dN[rsel];
            }
        }
        __syncthreads();
        if (tid == 0) {
            #pragma unroll 1
            for (int rsel = 0; rsel < 2; ++rsel) {
                double tSg = 0.0, tSg2 = 0.0; unsigned tN = 0u;
                for (int ww = 0; ww < nthr / 32; ++ww) {
                    tSg += wSg[rsel][ww]; tSg2 += wSg2[rsel][ww]; tN += wN[rsel][ww];
                }
                sN[rsel] += tN; sSg[rsel] += tSg; sSg2[rsel] += tSg2;

                const double n = (double)sN[rsel];
                const double mean = sSg[rsel] / fmax(n, 1.0);
                const double var = (sSg2[rsel] - 2.0 * mean * sSg[rsel] + n * mean * mean)
                                   / fmax(n - 1.0, 1.0);
                const double Dg = var + 0.15 * mean * mean;
                const double pl = 10.0 * sqrt(fmax(Dg, 1e-30));
                if (n >= 2.0) { sLoss += pl; sCnt += 1.0f; }   // step is valid here
            }
        }
        __syncthreads();
    }

    if (tid == 0) {
        const float depth_loss = (float)(sLoss / fmax((double)sCnt, 1.0));
        const float n_obj = fmaxf(det[0], 1.0f);
        const float no_object_loss = det[2] / fmaxf(det[1], 1.0f);
        const float object_loss = det[3] / n_obj;
        const float box_loss = det[4] / (n_obj * 4.0f);
        const float class_loss = det[5] / n_obj;
        out[0] = 10.0f * box_loss + object_loss + 10.0f * no_object_loss
               + class_loss + depth_loss;
    }
}

extern "C" void kernel_launch(void* const* d_in, const int* in_sizes, int n_in,
                              void* d_out, int out_size, void* d_ws, size_t ws_size,
                              hipStream_t stream) {
    (void)in_sizes; (void)n_in; (void)out_size; (void)ws_size;
    const float* pred = (const float*)d_in[0];
    const float* tgt  = (const float*)d_in[1];
    const float* anch = (const float*)d_in[2];
    const float* dtgt = (const float*)d_in[3];
    // d_in[4] is S (== 8), fixed by the problem shapes.

    float* wsdet = (float*)d_ws;                       // 6 detection scalars
    float* gp  = (float*)((char*)d_ws + 256);          // g-map for dp (2,256,256)
    float* gs  = gp + NPIX;                            // g-map for ds
    float* ltg = gs + NPIX;                            // precomputed log(target)

    det_loss_kernel<<<1, 512, 0, stream>>>(pred, tgt, anch, wsdet);
    depth_loss_kernel<<<1, 1024, 0, stream>>>(pred, anch, dtgt, gp, gs, ltg, wsdet,
                                              (float*)d_out);
}